// KANStressPredictor_85753317032317
// MI455X (gfx1250) — compile-verified
//
#include <hip/hip_runtime.h>

#define LN2f  0.69314718055994530942f
#define RLN2f 1.44269504088896340736f

#if __has_builtin(__builtin_amdgcn_global_load_async_to_lds_b128) && \
    __has_builtin(__builtin_amdgcn_s_wait_asynccnt)
#define KAN_USE_ASYNC_LDS 1
typedef int kan_v4i __attribute__((ext_vector_type(4)));
typedef __attribute__((address_space(1))) kan_v4i* kan_gp_t;
typedef __attribute__((address_space(3))) kan_v4i* kan_lp_t;
#endif

struct D4 { float v, x, y, z; };   // value + d/ds1, d/ds2, d/ds3

struct KParams {
  float c0[18];          // coef0 (3,1,6)
  float sb0[3], sp0[3];
  float b0;
  float c1[6];           // coef1 (1,1,6)
  float sb1, sp1, b1;
  float ki0, ki1;
};

__device__ __forceinline__ float frcp(float x)  { return __builtin_amdgcn_rcpf(x); }
__device__ __forceinline__ float frsq(float x)  { return __builtin_amdgcn_rsqf(x); }
__device__ __forceinline__ float flog2(float x) { return __builtin_amdgcn_logf(x); }   // v_log_f32
__device__ __forceinline__ float fexp2(float x) { return __builtin_amdgcn_exp2f(x); }  // v_exp_f32

// ---------------- reference-exact B-spline recursion (pre-kernel only) -------
__device__ __forceinline__ void bspline6(float x, const float* __restrict__ c,
                                         float& val, float& dval)
{
  const float g0 = -3.0f, h = 2.0f / 3.0f;
  float B0[9];
#pragma unroll
  for (int k = 0; k < 9; ++k) {
    float gk  = g0 + h * k;
    float gk1 = g0 + h * (k + 1);
    B0[k] = (x >= gk && x < gk1) ? 1.0f : 0.0f;
  }
  float B1[8];
#pragma unroll
  for (int k = 0; k < 8; ++k) {
    float u = (x - (g0 + h * k)) * 1.5f;
    float w = ((g0 + h * (k + 2)) - x) * 1.5f;
    B1[k] = u * B0[k] + w * B0[k + 1];
  }
  float B2[7];
#pragma unroll
  for (int k = 0; k < 7; ++k) {
    float u = (x - (g0 + h * k)) * 0.75f;
    float w = ((g0 + h * (k + 3)) - x) * 0.75f;
    B2[k] = u * B1[k] + w * B1[k + 1];
  }
  float B3[6];
#pragma unroll
  for (int k = 0; k < 6; ++k) {
    float u = (x - (g0 + h * k)) * 0.5f;
    float w = ((g0 + h * (k + 4)) - x) * 0.5f;
    B3[k] = u * B2[k] + w * B2[k + 1];
  }
  float v = 0.0f, dv = 0.0f;
#pragma unroll
  for (int k = 0; k < 6; ++k) {
    v  = fmaf(B3[k], c[k], v);
    dv = fmaf(1.5f * (B2[k] - B2[k + 1]), c[k], dv);
  }
  val = v; dval = dv;
}

__device__ __forceinline__ void silu_vd(float x, float& v, float& d)
{
  float e = fexp2(-x * RLN2f);
  float s = frcp(1.0f + e);
  v = x * s;
  d = s * (1.0f + x * (1.0f - s));
}

// Generic forward-mode point gradient using the recursion (pre-kernel / grad0).
__device__ __forceinline__ void point_grad_ref(float s1, float s2, float s3,
                                               const KParams& P, float out[3])
{
  float av = fmaf(2.0f, s1, 1.0f);
  float dv = fmaf(2.0f, s2, 1.0f);
  float bv = s3;

  D4 det = { av * dv - bv * bv, 2.0f * dv, 2.0f * av, -2.0f * bv };
  float rdet = frcp(det.v);
  float Ls = rdet * RLN2f;
  D4 L = { flog2(det.v), det.x * Ls, det.y * Ls, det.z * Ls };

  D4 m = { 0.5f * (av + dv), 1.0f, 1.0f, 0.0f };
  float ad = av - dv;
  D4 h2 = { 0.25f * ad * ad + bv * bv, ad, -ad, 2.0f * bv };

  bool pos = h2.v > 0.0f;
  float rs = frsq(pos ? h2.v : 1.0f);
  float rv = pos ? h2.v * rs : 0.0f;
  float rc = pos ? 0.5f * rs : 0.0f;
  D4 r = { rv, h2.x * rc, h2.y * rc, h2.z * rc };

  D4 e1 = { m.v - r.v, m.x - r.x, m.y - r.y, m.z - r.z };
  D4 e2 = { m.v + r.v, m.x + r.x, m.y + r.y, m.z + r.z };

  float l1 = flog2(e1.v), l2 = flog2(e2.v);
  float re1 = frcp(e1.v), re2 = frcp(e2.v);
  float k05 = 0.5f * RLN2f * P.ki0;
  float k16 = P.ki0 * (1.0f / 6.0f);

  float a1 = k05 * re1;
  D4 u1 = { P.ki0 * (0.5f * l1 - L.v * (1.0f / 6.0f)),
            a1 * e1.x - k16 * L.x, a1 * e1.y - k16 * L.y, a1 * e1.z - k16 * L.z };
  float t1v = fexp2(u1.v);
  float t1s = t1v * LN2f;
  D4 t1 = { t1v, u1.x * t1s, u1.y * t1s, u1.z * t1s };

  float a2 = k05 * re2;
  D4 u2 = { P.ki0 * (0.5f * l2 - L.v * (1.0f / 6.0f)),
            a2 * e2.x - k16 * L.x, a2 * e2.y - k16 * L.y, a2 * e2.z - k16 * L.z };
  float t2v = fexp2(u2.v);
  float t2s = t2v * LN2f;
  D4 t2 = { t2v, u2.x * t2s, u2.y * t2s, u2.z * t2s };

  float c3 = 0.5f * LN2f * P.ki1;
  D4 t3 = { c3 * L.v, c3 * L.x, c3 * L.y, c3 * L.z };

  D4 xin[3] = { t1, t2, t3 };
  D4 hsum = { P.b0, 0.0f, 0.0f, 0.0f };
#pragma unroll
  for (int i = 0; i < 3; ++i) {
    float sv, sd; silu_vd(xin[i].v, sv, sd);
    float pv, pd; bspline6(xin[i].v, &P.c0[6 * i], pv, pd);
    float fd = P.sb0[i] * sd + P.sp0[i] * pd;
    hsum.v += P.sb0[i] * sv + P.sp0[i] * pv;
    hsum.x = fmaf(fd, xin[i].x, hsum.x);
    hsum.y = fmaf(fd, xin[i].y, hsum.y);
    hsum.z = fmaf(fd, xin[i].z, hsum.z);
  }
  float sv, sd; silu_vd(hsum.v, sv, sd);
  float pv, pd; bspline6(hsum.v, P.c1, pv, pd);
  float wd = P.sb1 * sd + P.sp1 * pd;
  out[0] = wd * hsum.x;
  out[1] = wd * hsum.y;
  out[2] = wd * hsum.z;
}

__device__ __forceinline__ void load_params(KParams& P,
    const float* c0, const float* sb0, const float* sp0, const float* b0,
    const float* c1, const float* sb1, const float* sp1, const float* b1,
    const float* ki0, const float* ki1)
{
#pragma unroll
  for (int k = 0; k < 18; ++k) P.c0[k] = c0[k];
#pragma unroll
  for (int k = 0; k < 3; ++k) { P.sb0[k] = sb0[k]; P.sp0[k] = sp0[k]; }
  P.b0 = b0[0];
#pragma unroll
  for (int k = 0; k < 6; ++k) P.c1[k] = c1[k];
  P.sb1 = sb1[0]; P.sp1 = sp1[0]; P.b1 = b1[0];
  P.ki0 = ki0[0]; P.ki1 = ki1[0];
}

__device__ __forceinline__ float cwin(const float* c, int idx)
{
  return (idx >= 0 && idx < 6) ? c[idx] : 0.0f;
}

// Pre-kernel: grad at strain=0 -> ws[0..2]; piecewise-cubic monomial tables
// (9 intervals x float4, per channel, sp-scales folded) -> ws[4..147].
__global__ void kan_prep_kernel(
    const float* c0, const float* sb0, const float* sp0, const float* b0,
    const float* c1, const float* sb1, const float* sp1, const float* b1,
    const float* ki0, const float* ki1, float* __restrict__ ws)
{
  if (blockIdx.x != 0 || threadIdx.x != 0) return;
  KParams P;
  load_params(P, c0, sb0, sp0, b0, c1, sb1, sp1, b1, ki0, ki1);

  float o[3];
  point_grad_ref(0.0f, 0.0f, 0.0f, P, o);
  ws[0] = o[0]; ws[1] = o[1]; ws[2] = o[2]; ws[3] = 0.0f;

  float4* wt = (float4*)(ws + 4);
  for (int ch = 0; ch < 4; ++ch) {
    const float* c = (ch < 3) ? &P.c0[6 * ch] : P.c1;
    float sp = (ch < 3) ? P.sp0[ch] : P.sp1;
    for (int j = 0; j < 9; ++j) {
      float cm3 = cwin(c, j - 3), cm2 = cwin(c, j - 2);
      float cm1 = cwin(c, j - 1), cj  = cwin(c, j);
      float4 p;
      p.x = (cm3 + 4.0f * cm2 + cm1) * (1.0f / 6.0f) * sp;           // t^0
      p.y = (cm1 - cm3) * 0.5f * sp;                                  // t^1
      p.z = (cm3 - 2.0f * cm2 + cm1) * 0.5f * sp;                     // t^2
      p.w = (cj - cm3 + 3.0f * (cm2 - cm1)) * (1.0f / 6.0f) * sp;     // t^3
      wt[ch * 9 + j] = p;
    }
  }
}

// Fast per-channel spline value + d/dx from the LDS table.
__device__ __forceinline__ void chan_eval(float x, const float4* __restrict__ tab,
                                          int ch, float& val, float& dvx)
{
  float xs = fmaf(x, 1.5f, 4.5f);            // (x - g_0)/h with g_0 = -3, h = 2/3
  float fj = floorf(xs);
  fj = fminf(fmaxf(fj, 0.0f), 8.0f);
  float t = xs - fj;
  int j = (int)fj;
  float4 p = tab[ch * 9 + j];                // ds_load_b128
  float v  = fmaf(fmaf(fmaf(p.w, t, p.z), t, p.y), t, p.x);
  float dt = fmaf(fmaf(3.0f * p.w, t, 2.0f * p.z), t, p.y);
  bool in = (xs >= 0.0f) && (xs < 9.0f);
  val = in ? v : 0.0f;
  dvx = in ? dt * 1.5f : 0.0f;
}

// Fast point gradient: preproc duals + table-based KAN (sp folded into tables).
__device__ __forceinline__ void point_grad_fast(float s1, float s2, float s3,
    const float* __restrict__ sb0a, float b0v, float sb1v, float ki0v, float ki1v,
    const float4* __restrict__ tab, float out[3])
{
  float av = fmaf(2.0f, s1, 1.0f);
  float dv = fmaf(2.0f, s2, 1.0f);
  float bv = s3;

  D4 det = { av * dv - bv * bv, 2.0f * dv, 2.0f * av, -2.0f * bv };
  float rdet = frcp(det.v);
  float Ls = rdet * RLN2f;
  D4 L = { flog2(det.v), det.x * Ls, det.y * Ls, det.z * Ls };

  D4 m = { 0.5f * (av + dv), 1.0f, 1.0f, 0.0f };
  float ad = av - dv;
  D4 h2 = { 0.25f * ad * ad + bv * bv, ad, -ad, 2.0f * bv };

  bool pos = h2.v > 0.0f;
  float rs = frsq(pos ? h2.v : 1.0f);
  float rv = pos ? h2.v * rs : 0.0f;
  float rc = pos ? 0.5f * rs : 0.0f;
  D4 r = { rv, h2.x * rc, h2.y * rc, h2.z * rc };

  D4 e1 = { m.v - r.v, m.x - r.x, m.y - r.y, m.z - r.z };
  D4 e2 = { m.v + r.v, m.x + r.x, m.y + r.y, m.z + r.z };

  float l1 = flog2(e1.v), l2 = flog2(e2.v);
  float re1 = frcp(e1.v), re2 = frcp(e2.v);
  float k05 = 0.5f * RLN2f * ki0v;
  float k16 = ki0v * (1.0f / 6.0f);

  float a1 = k05 * re1;
  D4 u1 = { ki0v * (0.5f * l1 - L.v * (1.0f / 6.0f)),
            a1 * e1.x - k16 * L.x, a1 * e1.y - k16 * L.y, a1 * e1.z - k16 * L.z };
  float t1v = fexp2(u1.v);
  float t1s = t1v * LN2f;
  D4 t1 = { t1v, u1.x * t1s, u1.y * t1s, u1.z * t1s };

  float a2 = k05 * re2;
  D4 u2 = { ki0v * (0.5f * l2 - L.v * (1.0f / 6.0f)),
            a2 * e2.x - k16 * L.x, a2 * e2.y - k16 * L.y, a2 * e2.z - k16 * L.z };
  float t2v = fexp2(u2.v);
  float t2s = t2v * LN2f;
  D4 t2 = { t2v, u2.x * t2s, u2.y * t2s, u2.z * t2s };

  float c3 = 0.5f * LN2f * ki1v;
  D4 t3 = { c3 * L.v, c3 * L.x, c3 * L.y, c3 * L.z };

  D4 xin[3] = { t1, t2, t3 };
  D4 hsum = { b0v, 0.0f, 0.0f, 0.0f };
#pragma unroll
  for (int i = 0; i < 3; ++i) {
    float sv, sd; silu_vd(xin[i].v, sv, sd);
    float pv, pdx; chan_eval(xin[i].v, tab, i, pv, pdx);
    float fd = sb0a[i] * sd + pdx;           // sp already folded into table
    hsum.v += sb0a[i] * sv + pv;
    hsum.x = fmaf(fd, xin[i].x, hsum.x);
    hsum.y = fmaf(fd, xin[i].y, hsum.y);
    hsum.z = fmaf(fd, xin[i].z, hsum.z);
  }
  float sv, sd; silu_vd(hsum.v, sv, sd);
  float pv, pdx; chan_eval(hsum.v, tab, 3, pv, pdx);
  float wd = sb1v * sd + pdx;                // sp1 folded into table

  out[0] = wd * hsum.x;
  out[1] = wd * hsum.y;
  out[2] = wd * hsum.z;
}

// Main kernel: LDS table (async-staged when available), 4 points/thread,
// 3x float4 in / 3x float4 out, grid-stride + one-stride-ahead prefetch.
__global__ void kan_grad_kernel(
    const float* __restrict__ strain,
    const float* sb0, const float* b0, const float* sb1,
    const float* ki0, const float* ki1,
    const float* __restrict__ ws, float* __restrict__ out, int npts)
{
  __shared__ float4 s_tab[36];
  const float4* tabg = (const float4*)(ws + 4);
  int tix = threadIdx.x;
  if (tix < 36) {
#ifdef KAN_USE_ASYNC_LDS
    __builtin_amdgcn_global_load_async_to_lds_b128(
        (kan_gp_t)(tabg + tix), (kan_lp_t)(&s_tab[tix]), 0, 0);
#else
    s_tab[tix] = tabg[tix];
#endif
  }
#ifdef KAN_USE_ASYNC_LDS
  __builtin_amdgcn_s_wait_asynccnt(0);
#endif
  __syncthreads();

  float sb0a[3] = { sb0[0], sb0[1], sb0[2] };
  float b0v = b0[0], sb1v = sb1[0], ki0v = ki0[0], ki1v = ki1[0];
  float gx = ws[0], gy = ws[1], gz = ws[2];

  int n4 = npts >> 2;
  int stride = gridDim.x * blockDim.x;
  for (int q = blockIdx.x * blockDim.x + tix; q < n4; q += stride) {
    const float4* ip = (const float4*)strain + (size_t)q * 3;
    if (q + stride < n4)
      __builtin_prefetch((const float4*)strain + (size_t)(q + stride) * 3, 0, 1);
    float4 w0 = ip[0], w1 = ip[1], w2 = ip[2];

    float o[12];
    point_grad_fast(w0.x, w0.y, w0.z, sb0a, b0v, sb1v, ki0v, ki1v, s_tab, &o[0]);
    point_grad_fast(w0.w, w1.x, w1.y, sb0a, b0v, sb1v, ki0v, ki1v, s_tab, &o[3]);
    point_grad_fast(w1.z, w1.w, w2.x, sb0a, b0v, sb1v, ki0v, ki1v, s_tab, &o[6]);
    point_grad_fast(w2.y, w2.z, w2.w, sb0a, b0v, sb1v, ki0v, ki1v, s_tab, &o[9]);

    float4 r0 = { o[0] - gx, o[1]  - gy, o[2]  - gz, o[3]  - gx };
    float4 r1 = { o[4] - gy, o[5]  - gz, o[6]  - gx, o[7]  - gy };
    float4 r2 = { o[8] - gz, o[9]  - gx, o[10] - gy, o[11] - gz };
    float4* op = (float4*)out + (size_t)q * 3;
    op[0] = r0; op[1] = r1; op[2] = r2;
  }

  int tail = npts & 3;
  if (tail && blockIdx.x == 0 && tix < tail) {
    int p = (npts & ~3) + tix;
    float o[3];
    point_grad_fast(strain[p * 3 + 0], strain[p * 3 + 1], strain[p * 3 + 2],
                    sb0a, b0v, sb1v, ki0v, ki1v, s_tab, o);
    out[p * 3 + 0] = o[0] - gx;
    out[p * 3 + 1] = o[1] - gy;
    out[p * 3 + 2] = o[2] - gz;
  }
}

extern "C" void kernel_launch(void* const* d_in, const int* in_sizes, int n_in,
                              void* d_out, int out_size, void* d_ws, size_t ws_size,
                              hipStream_t stream)
{
  const float* strain = (const float*)d_in[0];
  const float* c0  = (const float*)d_in[1];
  const float* sb0 = (const float*)d_in[2];
  const float* sp0 = (const float*)d_in[3];
  const float* b0  = (const float*)d_in[4];
  const float* c1  = (const float*)d_in[5];
  const float* sb1 = (const float*)d_in[6];
  const float* sp1 = (const float*)d_in[7];
  const float* b1  = (const float*)d_in[8];
  const float* ki0 = (const float*)d_in[9];
  const float* ki1 = (const float*)d_in[10];
  float* out = (float*)d_out;
  float* ws  = (float*)d_ws;   // [0..2]=grad0, [4..147]=tables

  int npts = in_sizes[0] / 3;  // 4096*512 = 2,097,152

  kan_prep_kernel<<<1, 32, 0, stream>>>(c0, sb0, sp0, b0, c1, sb1, sp1, b1,
                                        ki0, ki1, ws);

  int n4 = npts >> 2;
  int blocks = (n4 + 255) / 256;
  if (blocks > 1024) blocks = 1024;
  if (blocks < 1) blocks = 1;
  kan_grad_kernel<<<blocks, 256, 0, stream>>>(strain, sb0, b0, sb1, ki0, ki1,
                                              ws, out, npts);
}